// CableIntegrator_17480516894908
// MI455X (gfx1250) — compile-verified
//
#include <hip/hip_runtime.h>
#include <stdint.h>

// ---------------------------------------------------------------------------
// Hodgkin-Huxley cable integrator for MI455X (gfx1250).
//   T=8192 sequential steps, N=2048 segments -> latency/sync-bound scan.
//   Single persistent workgroup (1024 thr = 32 wave32) on one WGP:
//     - V state double-buffered in LDS  -> ONE s_barrier per time step.
//     - Stimulus rows (8 KB) DMA'd by the Tensor Data Mover into a 4-deep
//       LDS ring, 3 steps ahead (tensor_load_to_lds, s_wait_tensorcnt<=2):
//       ~3 compute-steps (~1.5us) of HBM latency slack, fully overlapped.
//     - Streaming outputs written with nontemporal packed float2 stores.
// ---------------------------------------------------------------------------

#define T_STEPS 8192
#define NSEG    2048
#define BLOCK   1024   // 2 segments per thread
#define SDEPTH  4      // stimulus ring depth (power of two)

typedef uint32_t v4u __attribute__((ext_vector_type(4)));
typedef int32_t  v8i __attribute__((ext_vector_type(8)));
typedef int32_t  v4i __attribute__((ext_vector_type(4)));
typedef float    f2  __attribute__((ext_vector_type(2)));

// Low 32 bits of a flat pointer to __shared__ == LDS byte address
// (flat LDS aperture is {SHARED_BASE, offset}; HW truncates to addr[31:0]).
__device__ __forceinline__ uint32_t lds_addr_of(const void* p) {
  return (uint32_t)(uintptr_t)p;
}

// Issue one TDM load of a contiguous row of NSEG f32 from global into LDS.
// D# per CDNA5 ISA ch.8: group0 = {flags, lds_addr, global_addr, type=2},
// group1 = {data_size=4B, tensor_dim0=N, tensor_dim1=1, tile=(N,1)}.
// Higher-dim groups zero-filled (<=2D tensor).
__device__ __forceinline__ void tdm_load_row(const float* gsrc, uint32_t lds_byte) {
  uint64_t ga = (uint64_t)(uintptr_t)gsrc;
  v4u g0;
  g0[0] = 0x1u;                                             // count=1, user D#
  g0[1] = lds_byte;                                         // lds_addr
  g0[2] = (uint32_t)(ga & 0xFFFFFFFFu);                     // global_addr[31:0]
  g0[3] = (uint32_t)((ga >> 32) & 0x01FFFFFFu) | (2u << 30);// addr[56:32] | type=2
  v8i g1;
  g1[0] = 0x00020000;                 // workgroup_mask=0, data_size=2 (4B)
  g1[1] = (int)((uint32_t)NSEG << 16);// tensor_dim0[15:0]=2048 (abar_addr=0)
  g1[2] = (int)(1u << 16);            // tensor_dim0[31:16]=0, tensor_dim1=1
  g1[3] = (int)((uint32_t)NSEG << 16);// tensor_dim1 hi=0,  tile_dim0=2048
  g1[4] = 1;                          // tile_dim1=1, tile_dim2=0
  g1[5] = NSEG;                       // tensor_dim0_stride[31:0]
  g1[6] = 0;                          // stride hi / tensor_dim1_stride lo
  g1[7] = 0;
  v4i z4 = {0, 0, 0, 0};
  v8i z8 = {0, 0, 0, 0, 0, 0, 0, 0};
  __builtin_amdgcn_tensor_load_to_lds(g0, g1, z4, z4, z8, 0);
}

// One HH segment update. Returns new V and mem_cur; updates m,h,n in place.
__device__ __forceinline__ void hh_seg(float V, float Vm1, float Vp1,
                                       float gna, float gk, float gl, float gax,
                                       float istim,
                                       float& m, float& h, float& n,
                                       float& Vnew, float& memcur) {
  const float DT = 0.025f;
  float am = 0.1f * (V + 40.0f) / (1.0f - __expf(-(V + 40.0f) * 0.1f));
  float bm = 4.0f * __expf(-(V + 65.0f) * (1.0f / 18.0f));
  float ah = 0.07f * __expf(-(V + 65.0f) * 0.05f);
  float bh = 1.0f / (1.0f + __expf(-(V + 35.0f) * 0.1f));
  float an = 0.01f * (V + 55.0f) / (1.0f - __expf(-(V + 55.0f) * 0.1f));
  float bn = 0.125f * __expf(-(V + 65.0f) * (1.0f / 80.0f));

  float m3 = m * m * m;
  float n2 = n * n;
  float n4 = n2 * n2;
  float i_ion = gna * m3 * h * (V - 50.0f)
              + gk * n4 * (V + 77.0f)
              + gl * (V + 54.387f);
  float i_ax = gax * (Vm1 + Vp1 - 2.0f * V);

  memcur = i_ax + istim;                    // == CM*dV + i_ion exactly
  Vnew   = V + DT * (i_ax + istim - i_ion); // CM == 1
  m += DT * (am * (1.0f - m) - bm * m);
  h += DT * (ah * (1.0f - h) - bh * h);
  n += DT * (an * (1.0f - n) - bn * n);
}

__global__ __launch_bounds__(BLOCK) void hh_cable_kernel(
    const float* __restrict__ stim,  // T x N
    const float* __restrict__ V0,
    const float* __restrict__ m0,
    const float* __restrict__ h0,
    const float* __restrict__ n0,
    const float* __restrict__ gna_g,
    const float* __restrict__ gk_g,
    const float* __restrict__ gl_g,
    const float* __restrict__ gax_g,
    float* __restrict__ out_cur,     // T x N
    float* __restrict__ out_vlt,     // T x N
    float* __restrict__ out_Vf,
    float* __restrict__ out_mf,
    float* __restrict__ out_hf,
    float* __restrict__ out_nf) {
  __shared__ __align__(16) float Vlds[2][NSEG];        // double-buffered V
  __shared__ __align__(16) float Sbuf[SDEPTH][NSEG];   // stimulus ring

  const int tid = threadIdx.x;
  const int s0 = 2 * tid;
  const int s1 = s0 + 1;

  float V0r = V0[s0], V1r = V0[s1];
  float m0r = m0[s0], m1r = m0[s1];
  float h0r = h0[s0], h1r = h0[s1];
  float n0r = n0[s0], n1r = n0[s1];
  float gna0 = gna_g[s0], gna1 = gna_g[s1];
  float gk0 = gk_g[s0],   gk1 = gk_g[s1];
  float gl0 = gl_g[s0],   gl1 = gl_g[s1];
  float gax0 = gax_g[s0], gax1 = gax_g[s1];

  Vlds[0][s0] = V0r;
  Vlds[0][s1] = V1r;

  const uint32_t sbuf_base = lds_addr_of(&Sbuf[0][0]);

  // Prologue: DMA stimulus rows 0..SDEPTH-2 into the ring; ensure row 0 done.
  if (tid < 32) {
    for (int r = 0; r < SDEPTH - 1; ++r) {
      tdm_load_row(stim + (size_t)r * NSEG,
                   sbuf_base + (uint32_t)r * (NSEG * 4u));
    }
    __builtin_amdgcn_s_wait_tensorcnt(SDEPTH - 2);
  }
  __syncthreads();

  for (int t = 0; t < T_STEPS; ++t) {
    const int vc = t & 1;           // V read buffer
    const int sl = t & (SDEPTH - 1);// stimulus slot for this step

    // Kick off DMA of row t+SDEPTH-1 into its ring slot (overlapped).
    if (tid < 32) {
      const int pf = t + SDEPTH - 1;
      if (pf < T_STEPS) {
        tdm_load_row(stim + (size_t)pf * NSEG,
                     sbuf_base + (uint32_t)(pf & (SDEPTH - 1)) * (NSEG * 4u));
      }
    }

    // Sealed-end (zero-flux) neighbor voltages from the read buffer.
    float Vm1 = (tid == 0)         ? V0r : Vlds[vc][s0 - 1];
    float Vp1 = (tid == BLOCK - 1) ? V1r : Vlds[vc][s1 + 1];

    f2 st = *(const f2*)&Sbuf[sl][s0];

    float Vn0, Vn1, mc0, mc1;
    hh_seg(V0r, Vm1, V1r, gna0, gk0, gl0, gax0, st[0], m0r, h0r, n0r, Vn0, mc0);
    hh_seg(V1r, V0r, Vp1, gna1, gk1, gl1, gax1, st[1], m1r, h1r, n1r, Vn1, mc1);

    // Streaming outputs: mem_cur and pre-step V (nontemporal, packed b64).
    const size_t row = (size_t)t * NSEG + (size_t)s0;
    f2 c2; c2[0] = mc0; c2[1] = mc1;
    f2 v2; v2[0] = V0r; v2[1] = V1r;
    __builtin_nontemporal_store(c2, (f2*)(out_cur + row));
    __builtin_nontemporal_store(v2, (f2*)(out_vlt + row));

    // Publish new V into the OTHER buffer: no read/write conflict,
    // so a single barrier per step is sufficient.
    V0r = Vn0;
    V1r = Vn1;
    Vlds[vc ^ 1][s0] = V0r;
    Vlds[vc ^ 1][s1] = V1r;

    // In-order TDM completion: <= SDEPTH-2 outstanding guarantees the row
    // needed by step t+1 has landed in LDS.
    if (tid < 32) {
      __builtin_amdgcn_s_wait_tensorcnt(SDEPTH - 2);
    }
    __syncthreads();   // new V + next stimulus row visible to all waves
  }

  out_Vf[s0] = V0r; out_Vf[s1] = V1r;
  out_mf[s0] = m0r; out_mf[s1] = m1r;
  out_hf[s0] = h0r; out_hf[s1] = h1r;
  out_nf[s0] = n0r; out_nf[s1] = n1r;
}

extern "C" void kernel_launch(void* const* d_in, const int* in_sizes, int n_in,
                              void* d_out, int out_size, void* d_ws, size_t ws_size,
                              hipStream_t stream) {
  (void)in_sizes; (void)n_in; (void)out_size; (void)d_ws; (void)ws_size;

  const float* stim = (const float*)d_in[0];  // (T, N)
  const float* V0   = (const float*)d_in[1];
  const float* m0   = (const float*)d_in[2];
  const float* h0   = (const float*)d_in[3];
  const float* n0   = (const float*)d_in[4];
  const float* gna  = (const float*)d_in[5];
  const float* gk   = (const float*)d_in[6];
  const float* gl   = (const float*)d_in[7];
  const float* gax  = (const float*)d_in[8];

  float* out = (float*)d_out;
  float* out_cur = out;                                   // (T, N)
  float* out_vlt = out + (size_t)T_STEPS * NSEG;          // (T, N)
  float* fin     = out + 2 * (size_t)T_STEPS * NSEG;      // Vf, mf, hf, nf
  hh_cable_kernel<<<dim3(1), dim3(BLOCK), 0, stream>>>(
      stim, V0, m0, h0, n0, gna, gk, gl, gax,
      out_cur, out_vlt, fin, fin + NSEG, fin + 2 * NSEG, fin + 3 * NSEG);
}